// EaModel_4638564680402
// MI455X (gfx1250) — compile-verified
//
#include <hip/hip_runtime.h>
#include <hip/hip_bf16.h>

// ---------------------------------------------------------------------------
// EAGLE draft-tree generation for MI455X (gfx1250).
// fp32 skinny GEMMs done with V_WMMA_F32_16X16X4_F32 (16 rows padded, L<=10).
// Each wave computes two 16x16 output tiles; hot loop uses uniform 64-bit
// bases + 32-bit voffsets so loads select the saddr (GVS) form.
// ---------------------------------------------------------------------------

typedef float v2f __attribute__((ext_vector_type(2)));
typedef float v8f __attribute__((ext_vector_type(8)));

#define Dm 2048
#define Vv 32000
#define ATTN_STRIDE 1088
#define NEGVAL (-1000000000.0f)
#define SCALE 0.022097086912079608f  // 1/sqrt(2048)

// WMMA A-matrix packed layout for 32-bit A (16x4 per chunk):
// lane = row + ((k&2)?16:0), element = k&1, chunk = k>>2
__device__ __forceinline__ int apack_idx(int row, int k) {
  int lane = row + ((k & 2) ? 16 : 0);
  return ((k >> 2) << 6) + (lane << 1) + (k & 1);
}

// ---- pack [E[ids[r]], hs[r]] (16 x 4096, zero rows >= L) into A layout ----
__global__ void ea_pack_xin(const float* __restrict__ E, const int* __restrict__ ids,
                            const float* __restrict__ hs, int rep,
                            float* __restrict__ Ap, int L) {
  int idx = blockIdx.x * 256 + threadIdx.x;      // 16*4096 = 65536
  if (idx >= 16 * 4096) return;
  int row = idx >> 12;
  int k = idx & 4095;
  float v = 0.0f;
  if (row < L) {
    if (k < Dm) v = E[(size_t)ids[row] * Dm + k];
    else        v = hs[(size_t)(rep ? 0 : row) * Dm + (k - Dm)];
  }
  Ap[apack_idx(row, k)] = v;
}

// ---- pack row-major 16xK (zero rows >= L) into A layout ----
__global__ void ea_pack_rm(const float* __restrict__ src, float* __restrict__ Ap,
                           int K, int L) {
  int idx = blockIdx.x * 256 + threadIdx.x;      // 16*K
  if (idx >= 16 * K) return;
  int row = idx / K;
  int k = idx - row * K;
  float v = (row < L) ? src[(size_t)row * K + k] : 0.0f;
  Ap[apack_idx(row, k)] = v;
}

// ---- C[16xN] = Apacked[16xK] @ B[KxN] (B row-major), fp32 WMMA 16x16x4 ----
// One wave -> two adjacent 16-col tiles (32 cols); block (8 waves) -> 256 cols.
__global__ void ea_gemm_wmma(const float* __restrict__ Ap, const float* __restrict__ B,
                             float* __restrict__ C, int K, int N,
                             int storeRows, int outStride) {
  int lane = threadIdx.x & 31;
  int wave = threadIdx.x >> 5;
  int n0 = (blockIdx.x * 8 + wave) * 32;
  if (n0 >= N) return;                           // wave-uniform: EXEC stays full
  int col = n0 + (lane & 15);
  int rsel = (lane < 16) ? 0 : 2;                // lanes 0-15: K rows 0,1; 16-31: rows 2,3

  const unsigned uN = (unsigned)N;
  const unsigned stepB = 4u * uN;                // 4 K-rows per chunk (elements)
  unsigned off0 = (unsigned)rsel * uN + (unsigned)col;  // tile 0, row k
  unsigned off1 = off0 + 16u;                           // tile 1
  unsigned offA = (unsigned)lane;                       // packed A in v2f units
  const v2f* Ap2 = (const v2f*)Ap;

  v8f acc0 = {0.f, 0.f, 0.f, 0.f, 0.f, 0.f, 0.f, 0.f};
  v8f acc1 = acc0;
  int kchunks = K >> 2;
#pragma unroll 4
  for (int kc = 0; kc < kchunks; ++kc) {
    v2f a = Ap2[offA];
    offA += 32u;
    v2f b0, b1;
    b0[0] = B[off0];
    b0[1] = B[off0 + uN];
    b1[0] = B[off1];
    b1[1] = B[off1 + uN];
    if (kc + 8 < kchunks) {
      __builtin_prefetch(B + off0 + 8u * stepB, 0, 1);
      __builtin_prefetch(B + off1 + 8u * stepB, 0, 1);
    }
    off0 += stepB;
    off1 += stepB;
    acc0 = __builtin_amdgcn_wmma_f32_16x16x4_f32(false, a, false, b0,
                                                 (short)0, acc0, false, false);
    acc1 = __builtin_amdgcn_wmma_f32_16x16x4_f32(false, a, false, b1,
                                                 (short)0, acc1, false, false);
  }
  // C/D layout: VGPR j holds row (j + (lane<16?0:8)), col tile_base + lane%16
  int rbase = (lane < 16) ? 0 : 8;
#pragma unroll
  for (int j = 0; j < 8; ++j) {
    int row = rbase + j;
    if (row < storeRows) {
      C[(size_t)row * outStride + col] = acc0[j];
      C[(size_t)row * outStride + col + 16] = acc1[j];
    }
  }
}

// ---- scores + masked softmax, one block per query row ----
__global__ void ea_attn_scores(const float* __restrict__ q_rm, const float* __restrict__ Kc,
                               float* __restrict__ attn, int pos0, int kv_hi) {
  int r = blockIdx.x;
  __shared__ float q_s[Dm];
  __shared__ float s_s[ATTN_STRIDE];
  __shared__ float red[8];
  int tid = threadIdx.x;                         // 256 threads, 8 waves
  int lane = tid & 31, wave = tid >> 5;
  for (int i = tid; i < Dm; i += 256) q_s[i] = q_rm[(size_t)r * Dm + i];
  __syncthreads();
  for (int c = wave; c < kv_hi; c += 8) {
    const float* krow = Kc + (size_t)c * Dm;
    float acc = 0.0f;
    for (int i = lane; i < Dm; i += 32) acc += q_s[i] * krow[i];
#pragma unroll
    for (int o = 16; o > 0; o >>= 1) acc += __shfl_xor(acc, o, 32);
    if (lane == 0) {
      bool ok = (c <= pos0 + r) && (c < 1024 || ((c - 1024) % 10) == r);
      s_s[c] = ok ? acc * SCALE : NEGVAL;
    }
  }
  __syncthreads();
  // softmax over s_s[0..kv_hi)
  float m = -INFINITY;
  for (int c = tid; c < kv_hi; c += 256) m = fmaxf(m, s_s[c]);
#pragma unroll
  for (int o = 16; o > 0; o >>= 1) m = fmaxf(m, __shfl_xor(m, o, 32));
  if (lane == 0) red[wave] = m;
  __syncthreads();
  if (tid == 0) {
    float mm = red[0];
    for (int w = 1; w < 8; ++w) mm = fmaxf(mm, red[w]);
    red[0] = mm;
  }
  __syncthreads();
  m = red[0];
  __syncthreads();
  float p = 0.0f;
  for (int c = tid; c < kv_hi; c += 256) {
    float e = expf(s_s[c] - m);
    s_s[c] = e;
    p += e;
  }
#pragma unroll
  for (int o = 16; o > 0; o >>= 1) p += __shfl_xor(p, o, 32);
  if (lane == 0) red[wave] = p;
  __syncthreads();
  if (tid == 0) {
    float pp = 0.0f;
    for (int w = 0; w < 8; ++w) pp += red[w];
    red[0] = pp;
  }
  __syncthreads();
  float inv = 1.0f / red[0];
  for (int c = tid; c < kv_hi; c += 256)
    attn[(size_t)r * ATTN_STRIDE + c] = s_s[c] * inv;
}

// ---- h[L x 2048] = attn[L x kv_hi] @ Vc[kv_hi x 2048]; one block per 128 cols ----
__global__ void ea_attn_v(const float* __restrict__ attn, const float* __restrict__ Vc,
                          float* __restrict__ h_rm, int L, int kv_hi) {
  __shared__ float a_s[10 * ATTN_STRIDE];        // 43.5 KB
  int tid = threadIdx.x;                         // 128
  int d = blockIdx.x * 128 + tid;
  for (int i = tid; i < L * ATTN_STRIDE; i += 128) a_s[i] = attn[i];
  __syncthreads();
  float acc[10];
#pragma unroll
  for (int r = 0; r < 10; ++r) acc[r] = 0.0f;
  for (int c = 0; c < kv_hi; ++c) {
    float v = Vc[(size_t)c * Dm + d];
#pragma unroll
    for (int r = 0; r < 10; ++r)
      if (r < L) acc[r] += a_s[r * ATTN_STRIDE + c] * v;
  }
#pragma unroll
  for (int r = 0; r < 10; ++r)
    if (r < L) h_rm[(size_t)r * Dm + d] = acc[r];
}

// ---- deterministic top-10 per row (ties -> lower index), writes out + next ids ----
__global__ void ea_topk10(const float* __restrict__ logits,
                          int* __restrict__ out_tok, float* __restrict__ out_prob,
                          int out_base, int* __restrict__ ids_next, int mode) {
  int r = blockIdx.x;
  const float* row = logits + (size_t)r * Vv;
  __shared__ int found[10];
  __shared__ float rv[32];
  __shared__ int ri[32];
  int tid = threadIdx.x;                         // 1024
  int lane = tid & 31, wave = tid >> 5;
  for (int pass = 0; pass < 10; ++pass) {
    float bv = -INFINITY;
    int bi = 0x7fffffff;
    for (int e = tid; e < Vv; e += 1024) {
      float v = row[e];
      bool excl = false;
      for (int f = 0; f < pass; ++f) excl |= (found[f] == e);
      if (!excl && (v > bv || (v == bv && e < bi))) { bv = v; bi = e; }
    }
#pragma unroll
    for (int o = 16; o > 0; o >>= 1) {
      float ov = __shfl_xor(bv, o, 32);
      int oi = __shfl_xor(bi, o, 32);
      if (ov > bv || (ov == bv && oi < bi)) { bv = ov; bi = oi; }
    }
    if (lane == 0) { rv[wave] = bv; ri[wave] = bi; }
    __syncthreads();
    if (wave == 0) {
      bv = rv[lane];
      bi = ri[lane];
#pragma unroll
      for (int o = 16; o > 0; o >>= 1) {
        float ov = __shfl_xor(bv, o, 32);
        int oi = __shfl_xor(bi, o, 32);
        if (ov > bv || (ov == bv && oi < bi)) { bv = ov; bi = oi; }
      }
      if (lane == 0) {
        found[pass] = bi;
        out_tok[(out_base + r) * 10 + pass] = bi;
        out_prob[(out_base + r) * 10 + pass] = bv;
        if (mode == 0) ids_next[pass] = bi;            // level 0: 10 tokens of row 0
        else if (mode == 1 && pass == 0) ids_next[r] = bi;  // argmax of each row
      }
    }
    __syncthreads();
  }
}

// ---------------------------------------------------------------------------
extern "C" void kernel_launch(void* const* d_in, const int* in_sizes, int n_in,
                              void* d_out, int out_size, void* d_ws, size_t ws_size,
                              hipStream_t stream) {
  const float* hidden_states = (const float*)d_in[0];
  const int*   input_ids     = (const int*)d_in[1];
  const float* E     = (const float*)d_in[2];
  const float* W_fc  = (const float*)d_in[3];
  const float* Wq    = (const float*)d_in[4];
  const float* Wk    = (const float*)d_in[5];
  const float* Wv    = (const float*)d_in[6];
  const float* Wo    = (const float*)d_in[7];
  const float* W_lm  = (const float*)d_in[8];
  const float* K_in  = (const float*)d_in[9];
  const float* V_in  = (const float*)d_in[10];

  // workspace carve-up
  float* ws = (float*)d_ws;
  size_t off = 0;
  auto alloc = [&](size_t n) { float* p = ws + off; off += (n + 63) & ~(size_t)63; return p; };
  float* Kc     = alloc((size_t)2048 * Dm);
  float* Vc     = alloc((size_t)2048 * Dm);
  float* Ap_big = alloc((size_t)16 * 4096);   // A-pack for W_fc (also reused for Wo)
  float* Ap_sm  = alloc((size_t)16 * Dm);     // A-pack for q/k/v and W_lm
  float* x_rm   = alloc((size_t)16 * Dm);
  float* q_rm   = alloc((size_t)16 * Dm);
  float* h_rm   = alloc((size_t)16 * Dm);
  float* h2_rm  = alloc((size_t)16 * Dm);
  float* logits = alloc((size_t)16 * Vv);
  float* attn   = alloc((size_t)16 * ATTN_STRIDE);
  int*   ids_ws = (int*)alloc(64);

  // functional KV-cache update: copy inputs to workspace (inputs stay immutable)
  (void)hipMemcpyAsync(Kc, K_in, sizeof(float) * 2048 * Dm, hipMemcpyDeviceToDevice, stream);
  (void)hipMemcpyAsync(Vc, V_in, sizeof(float) * 2048 * Dm, hipMemcpyDeviceToDevice, stream);

  int* out_tok = (int*)d_out;                 // 51*10 int32 tokens
  float* out_prob = (float*)d_out + 510;      // 51*10 fp32 probs

  for (int s = 0; s <= 5; ++s) {
    const int L = (s == 0) ? 1 : 10;
    const int pos0 = (s == 0) ? 1023 : 1024 + 10 * (s - 1);
    const int kv_hi = (s == 0) ? 1024 : pos0 + 10;
    const int rep = (s == 1) ? 1 : 0;
    const int out_base = (s == 0) ? 0 : 1 + 10 * (s - 1);
    const int mode = (s == 0) ? 0 : ((s < 5) ? 1 : 2);
    const int* idsPtr = (s == 0) ? (input_ids + 1024) : ids_ws;
    const float* hsPtr = (s == 0) ? hidden_states : h2_rm;

    // x = [E[ids], hs] @ W_fc
    ea_pack_xin<<<256, 256, 0, stream>>>(E, idsPtr, hsPtr, rep, Ap_big, L);
    ea_gemm_wmma<<<8, 256, 0, stream>>>(Ap_big, W_fc, x_rm, 4096, Dm, 16, Dm);
    ea_pack_rm<<<128, 256, 0, stream>>>(x_rm, Ap_sm, Dm, 16);
    // q / k / v (k,v scattered straight into the cache rows)
    ea_gemm_wmma<<<8, 256, 0, stream>>>(Ap_sm, Wq, q_rm, Dm, Dm, 16, Dm);
    ea_gemm_wmma<<<8, 256, 0, stream>>>(Ap_sm, Wk, Kc + (size_t)pos0 * Dm, Dm, Dm, L, Dm);
    ea_gemm_wmma<<<8, 256, 0, stream>>>(Ap_sm, Wv, Vc + (size_t)pos0 * Dm, Dm, Dm, L, Dm);
    // attention
    ea_attn_scores<<<L, 256, 0, stream>>>(q_rm, Kc, attn, pos0, kv_hi);
    ea_attn_v<<<16, 128, 0, stream>>>(attn, Vc, h_rm, L, kv_hi);
    // h2 = h @ Wo
    ea_pack_rm<<<128, 256, 0, stream>>>(h_rm, Ap_big, Dm, L);
    ea_gemm_wmma<<<8, 256, 0, stream>>>(Ap_big, Wo, h2_rm, Dm, Dm, 16, Dm);
    // logits = h2 @ W_lm  (dominant: 262 MB weight stream per step)
    ea_pack_rm<<<128, 256, 0, stream>>>(h2_rm, Ap_sm, Dm, L);
    ea_gemm_wmma<<<125, 256, 0, stream>>>(Ap_sm, W_lm, logits, Dm, Vv, 16, Vv);
    // top-10 per row, also produces next step's token ids
    ea_topk10<<<L, 1024, 0, stream>>>(logits, out_tok, out_prob, out_base, ids_ws, mode);
  }
}